// BondGINEStack_50929722196749
// MI455X (gfx1250) — compile-verified
//
#include <hip/hip_runtime.h>

// GINEConv stack, 3 layers, N=50000 nodes, E=800000 edges, H=128, ED=64.
// fp32 end-to-end using V_WMMA_F32_16X16X4_F32 (memory-bound workload; fp32
// WMMA keeps reference precision at zero bandwidth cost).

#define NN 50000
#define NE 800000
#define HH 128
#define EDIM 64
#define NLAYER 3
#define LN_EPS 1e-5f

#define ETILES (NE / 16)   // 50000 edge tiles
#define NTILES (NN / 16)   // 3125 node tiles
#define EBLOCKS 800        // persistent edge blocks (8 waves each)
#define WE_STRIDE 68       // LDS pad: bank = (4*col + k) % 64, conflict-free

typedef float v2f __attribute__((ext_vector_type(2)));
typedef float v8f __attribute__((ext_vector_type(8)));

__device__ __forceinline__ v8f wmma4(v2f a, v2f b, v8f c) {
  // 8 args: (neg_a, A, neg_b, B, c_mod, C, reuse_a, reuse_b)
  return __builtin_amdgcn_wmma_f32_16x16x4_f32(false, a, false, b, (short)0, c,
                                               false, false);
}

// ---------------------------------------------------------------------------
// Edge kernel (persistent, grid-stride over 16-edge tiles).
//   Stage We transposed into LDS once per block (B frag = one 8B ds_load).
//   e   = edge_attr[tile] @ We + be            (WMMA f32 16x16x4, K-loop=16)
//   msg = relu(s[src] + e)
//   agg[dst] += msg                            (global f32 atomics, L2-resident)
// Gathers for each n-tile are issued BEFORE that n-tile's WMMA chain (latency
// hidden under 16 WMMAs + 16 DS loads); the n-tile loop is kept rolled
// (#pragma unroll 1) to cap register pressure < 256 and preserve occupancy.
// ---------------------------------------------------------------------------
__global__ __launch_bounds__(256)
void gine_edge_kernel(const float* __restrict__ s,
                      const int* __restrict__ src,
                      const int* __restrict__ dst,
                      const float* __restrict__ ea,   // [E, 64]
                      const float* __restrict__ We,   // [64, 128]
                      const float* __restrict__ be,   // [128]
                      float* __restrict__ agg)        // [N, 128], pre-zeroed
{
  __shared__ __align__(16) float weT[HH * WE_STRIDE];  // WeT[col][k], 34 KB

  // Stage We transposed: coalesced global read, once per block.
  for (int idx = threadIdx.x; idx < EDIM * HH; idx += 256) {
    const int k   = idx >> 7;        // row of We
    const int col = idx & 127;       // col of We
    weT[col * WE_STRIDE + k] = We[idx];
  }
  __syncthreads();

  const int wav  = threadIdx.x >> 5;
  const int lane = threadIdx.x & 31;
  const int m    = lane & 15;                     // row (A) / col (B,C)
  const int kh   = lane >> 4;                     // K-half select
  const int nwaves = EBLOCKS * 8;

  for (int tile = blockIdx.x * 8 + wav; tile < ETILES; tile += nwaves) {
    const int e0 = tile * 16;

    // Prefetch next tile's streams (edge_attr row + endpoint indices).
    {
      const int nxt = tile + nwaves;
      if (nxt < ETILES) {
        __builtin_prefetch(ea + (size_t)(nxt * 16 + m) * EDIM + 2 * kh, 0, 3);
        __builtin_prefetch(src + nxt * 16, 0, 3);
        __builtin_prefetch(dst + nxt * 16, 0, 3);
      }
    }

    // A fragments: lane holds edge_attr[e0+m][4i+2kh .. +1]
    v2f a[16];
    const float* arow = ea + (size_t)(e0 + m) * EDIM + 2 * kh;
#pragma unroll
    for (int i = 0; i < 16; ++i) a[i] = *(const v2f*)(arow + 4 * i);

    // Edge endpoints for this tile's 16 rows.
    int jsb[8], jdb[8];
#pragma unroll
    for (int v = 0; v < 8; ++v) {
      const int eid = e0 + v + 8 * kh;
      jsb[v] = src[eid] * HH;                     // broadcast within half-wave
      jdb[v] = dst[eid] * HH;
    }

#pragma unroll 1   // rolled: bounded VGPR pressure, no multi-buffered accs
    for (int nt = 0; nt < 8; ++nt) {
      const int col = nt * 16 + m;

      // Issue this n-tile's 8 gathers first; consumed only after the WMMAs.
      float svl[8];
#pragma unroll
      for (int v = 0; v < 8; ++v)
        svl[v] = s[(size_t)(jsb[v] + col)];       // coalesced per half-wave
      const float bias = be[col];

      const float* bcol = &weT[col * WE_STRIDE + 2 * kh];
      v8f c = {};
#pragma unroll
      for (int i = 0; i < 16; ++i) {
        v2f b = *(const v2f*)(bcol + 4 * i);      // single conflict-free ds_b64
        c = wmma4(a[i], b, c);
      }

#pragma unroll
      for (int v = 0; v < 8; ++v) {
        float msg = svl[v] + c[v] + bias;
        msg = msg > 0.f ? msg : 0.f;
        atomicAdd(&agg[(size_t)(jdb[v] + col)], msg);  // no-return, STOREcnt
      }
    }
  }
}

// ---------------------------------------------------------------------------
// Node kernel: one wave per 16-node tile.
//   h  = s + agg
//   t  = silu(h @ W1 + b1)      (WMMA; t staged via LDS for layout transpose)
//   u  = t @ W2 + b2            (WMMA, A read back from LDS in A-layout)
//   x  = s + u;  LayerNorm(x) via shfl_xor row reduction;  out = silu(.)
// ---------------------------------------------------------------------------
__global__ __launch_bounds__(128)
void gine_node_kernel(const float* __restrict__ s_in,
                      const float* __restrict__ agg,
                      const float* __restrict__ W1,
                      const float* __restrict__ b1,
                      const float* __restrict__ W2,
                      const float* __restrict__ b2,
                      const float* __restrict__ gam,
                      const float* __restrict__ bet,
                      float* __restrict__ s_out)
{
  __shared__ __align__(16) float t_all[4][16][132];  // stride 132: conflict-free
  const int wav  = threadIdx.x >> 5;
  const int lane = threadIdx.x & 31;
  const int tile = blockIdx.x * 4 + wav;
  if (tile >= NTILES) return;                        // wave-uniform exit
  float (*t_lds)[132] = t_all[wav];

  const int m    = lane & 15;
  const int kh   = lane >> 4;
  const int row0 = tile * 16;

  // h = s + agg, preloaded in A-layout (32 contiguous float2 per lane)
  v2f a[32];
  {
    const float* sr = s_in + (size_t)(row0 + m) * HH + 2 * kh;
    const float* ar = agg  + (size_t)(row0 + m) * HH + 2 * kh;
#pragma unroll
    for (int i = 0; i < 32; ++i) {
      v2f sv = *(const v2f*)(sr + 4 * i);
      v2f av = *(const v2f*)(ar + 4 * i);
      a[i] = sv + av;
    }
  }

  // GEMM1: t = silu(h @ W1 + b1) -> LDS (C-layout write, A-layout read later)
  for (int nt = 0; nt < 8; ++nt) {
    const int col = nt * 16 + m;
    v8f c = {};
#pragma unroll
    for (int i = 0; i < 32; ++i) {
      const int k = 4 * i + 2 * kh;
      v2f b;
      b.x = W1[(size_t)k * HH + col];
      b.y = W1[(size_t)(k + 1) * HH + col];
      c = wmma4(a[i], b, c);
    }
    const float bias = b1[col];
#pragma unroll
    for (int v = 0; v < 8; ++v) {
      float y = c[v] + bias;
      y = y / (1.f + __expf(-y));                  // silu
      t_lds[v + 8 * kh][col] = y;
    }
  }
  // Same-wave LDS RAW: DS pipeline is in-order per wave; compiler inserts
  // s_wait_dscnt for the dependent loads below. No block barrier needed.

  // GEMM2: u = t @ W2 (A fragments re-read from LDS as contiguous float2)
  v8f u[8];
  for (int nt = 0; nt < 8; ++nt) {
    const int col = nt * 16 + m;
    v8f c = {};
#pragma unroll
    for (int i = 0; i < 32; ++i) {
      const int k = 4 * i + 2 * kh;
      v2f av = *(const v2f*)&t_lds[m][k];
      v2f b;
      b.x = W2[(size_t)k * HH + col];
      b.y = W2[(size_t)(k + 1) * HH + col];
      c = wmma4(av, b, c);
    }
    u[nt] = c;
  }

  // Epilogue: x = s + u + b2 ; LayerNorm over H=128 ; silu ; store
  float x[8][8];
  float rsum[8], rsq[8];
#pragma unroll
  for (int v = 0; v < 8; ++v) { rsum[v] = 0.f; rsq[v] = 0.f; }
#pragma unroll
  for (int nt = 0; nt < 8; ++nt) {
    const int col  = nt * 16 + m;
    const float bias = b2[col];
#pragma unroll
    for (int v = 0; v < 8; ++v) {
      const int row = row0 + v + 8 * kh;
      float val = s_in[(size_t)row * HH + col] + u[nt][v] + bias;
      x[v][nt] = val;
      rsum[v] += val;
      rsq[v]  += val * val;
    }
  }
  // Row reduction across the 16 lanes of each half-wave (wave32 shuffles;
  // masks 1..8 never cross the kh boundary).
#pragma unroll
  for (int v = 0; v < 8; ++v) {
#pragma unroll
    for (int msk = 1; msk < 16; msk <<= 1) {
      rsum[v] += __shfl_xor(rsum[v], msk, 32);
      rsq[v]  += __shfl_xor(rsq[v],  msk, 32);
    }
  }
#pragma unroll
  for (int nt = 0; nt < 8; ++nt) {
    const int col = nt * 16 + m;
    const float g  = gam[col];
    const float bb = bet[col];
#pragma unroll
    for (int v = 0; v < 8; ++v) {
      const float mu  = rsum[v] * (1.f / 128.f);
      const float var = rsq[v] * (1.f / 128.f) - mu * mu;
      float y = (x[v][nt] - mu) * rsqrtf(var + LN_EPS) * g + bb;
      y = y / (1.f + __expf(-y));                  // silu
      s_out[(size_t)(row0 + v + 8 * kh) * HH + col] = y;
    }
  }
}

// ---------------------------------------------------------------------------
extern "C" void kernel_launch(void* const* d_in, const int* in_sizes, int n_in,
                              void* d_out, int out_size, void* d_ws,
                              size_t ws_size, hipStream_t stream) {
  (void)in_sizes; (void)n_in; (void)out_size; (void)ws_size;
  const float* s0    = (const float*)d_in[0];
  const int*   eidx  = (const int*)d_in[1];
  const float* eattr = (const float*)d_in[2];
  const float* We    = (const float*)d_in[3];
  const float* be    = (const float*)d_in[4];
  const float* W1    = (const float*)d_in[5];
  const float* b1    = (const float*)d_in[6];
  const float* W2    = (const float*)d_in[7];
  const float* b2    = (const float*)d_in[8];
  const float* gam   = (const float*)d_in[9];
  const float* bet   = (const float*)d_in[10];

  float* out   = (float*)d_out;
  float* s_buf = (float*)d_ws;                       // [N,H]
  float* agg   = s_buf + (size_t)NN * HH;            // [N,H]
  const int* src = eidx;
  const int* dst = eidx + NE;

  const float* cur = s0;
  for (int l = 0; l < NLAYER; ++l) {
    hipMemsetAsync(agg, 0, (size_t)NN * HH * sizeof(float), stream);
    gine_edge_kernel<<<EBLOCKS, 256, 0, stream>>>(
        cur, src, dst, eattr, We + (size_t)l * EDIM * HH, be + l * HH, agg);
    float* s_next = (l == NLAYER - 1) ? out : s_buf;
    gine_node_kernel<<<(NTILES + 3) / 4, 128, 0, stream>>>(
        cur, agg, W1 + (size_t)l * HH * HH, b1 + l * HH,
        W2 + (size_t)l * HH * HH, b2 + l * HH, gam + l * HH, bet + l * HH,
        s_next);
    cur = s_next;
  }
}